// FastWeights_85753317032517
// MI455X (gfx1250) — compile-verified
//
#include <hip/hip_runtime.h>
#include <stdint.h>

// FastWeights RNN (diagonal fast-weight variant), fp32 end-to-end.
//   1) xW[T*B, H]  = input[T*B, I] @ W_ih[I, H]     (WMMA f32 16x16x4, LDS-tiled,
//                                                    async global->LDS double buffer)
//   2) elementwise 256-step recurrence -> h_last     (VALU, L2-resident xW)
//   3) out[B, O]   = h_last @ W_y + b_y              (same WMMA kernel)
// Workspace: xW (T*B*H floats = 128MB), then h_last (B*H floats).

typedef __attribute__((ext_vector_type(2))) float v2f;
typedef __attribute__((ext_vector_type(8))) float v8f;

#define T_DIM 256
#define B_DIM 128
#define I_DIM 512
#define H_DIM 1024
#define O_DIM 512

#define FAST_LR 0.5f
#define DECAY_LR 0.95f

#define BK 32         // K-chunk staged per double-buffer step
#define SA_STRIDE 36  // floats: 144B rows -> 16B aligned; 36*r mod 64 distinct for r<16
#define SB_STRIDE 260 // floats: 1040B rows -> 16B aligned; rows k,k+2 are 8 banks apart

// CDNA5 async global->LDS copy (ASYNCcnt-tracked); per-lane 16B move.
__device__ __forceinline__ void async_g2l_b128(uint32_t lds_byte, const void* gaddr) {
  asm volatile("global_load_async_to_lds_b128 %0, %1, off"
               :: "v"(lds_byte), "v"(gaddr)
               : "memory");
}
__device__ __forceinline__ void wait_async0() {
  asm volatile("s_wait_asynccnt 0x0" ::: "memory");
}

// ---------------------------------------------------------------------------
// fp32 WMMA GEMM: C[M,N] = A[M,K] * B[K,N] (+ bias[N] if non-null).
// Block = 256 threads = 8 waves (2M x 4N), block tile 32(M) x 256(N).
// Grid: (M/32, N/256). Requires M%32==0, N%256==0, K%BK==0.
//
// fp32 WMMA fragment layouts (CDNA5 ISA 7.12.2), half = lane>>4, l = lane&15:
//   A 16x4 : v0 = A[l, k0+2*half], v1 = A[l, k0+2*half+1]
//   B 4x16 : v0 = B[k0+2*half, col=l], v1 = B[k0+2*half+1, col=l]
//   C 16x16: vr = C[row_base + r + 8*half, col_base + l]
// ---------------------------------------------------------------------------
__global__ __launch_bounds__(256) void fw_gemm_wmma(
    const float* __restrict__ A, const float* __restrict__ Bm,
    const float* __restrict__ bias, float* __restrict__ C,
    int M, int K, int N) {
  __shared__ __align__(16) float sA[2][32 * SA_STRIDE];
  __shared__ __align__(16) float sB[2][BK * SB_STRIDE];

  const int tid  = threadIdx.x;
  const int lane = tid & 31;
  const int wave = tid >> 5;
  const int l    = lane & 15;
  const int half = lane >> 4;
  const int wm   = wave >> 2;  // 0..1
  const int wn   = wave & 3;   // 0..3

  const int row_blk = blockIdx.x * 32;
  const int col_blk = blockIdx.y * 256;

  // Cooperative staging assignment (256 threads):
  //   A tile 32x32 floats : 1 x b128 per thread.
  //   B tile 32x256 floats: 8 x b128 per thread.
  const int a_row = tid >> 3;       // 0..31
  const int a_c4  = (tid & 7) * 4;  // 0,4,..,28
  const uint32_t sA_base = (uint32_t)(uintptr_t)&sA[0][0];
  const uint32_t sB_base = (uint32_t)(uintptr_t)&sB[0][0];

  auto stage = [&](int kc, int buf) {
    async_g2l_b128(
        sA_base + (uint32_t)((buf * 32 + a_row) * SA_STRIDE + a_c4) * 4u,
        A + (size_t)(row_blk + a_row) * K + kc + a_c4);
#pragma unroll
    for (int i = 0; i < 8; ++i) {
      const int flat = i * 256 + tid;     // float4 slot 0..2047
      const int br   = flat >> 6;         // 0..31
      const int bc4  = (flat & 63) * 4;   // 0..252
      async_g2l_b128(
          sB_base + (uint32_t)((buf * BK + br) * SB_STRIDE + bc4) * 4u,
          Bm + (size_t)(kc + br) * N + col_blk + bc4);
    }
  };

  v8f acc[4];
  acc[0] = v8f{}; acc[1] = v8f{}; acc[2] = v8f{}; acc[3] = v8f{};

  // Prologue: fill buffer 0.
  stage(0, 0);
  wait_async0();
  __syncthreads();

  const int nchunk = K / BK;
  for (int kk = 0; kk < nchunk; ++kk) {
    const int cur = kk & 1;
    if (kk + 1 < nchunk) stage((kk + 1) * BK, cur ^ 1);

    const float* __restrict__ sa = &sA[cur][(16 * wm + l) * SA_STRIDE];
    const float* __restrict__ sb = &sB[cur][64 * wn + l];

#pragma unroll
    for (int k0 = 0; k0 < BK; k0 += 4) {
      const int ka = k0 + 2 * half;
      v2f a;
      a.x = sa[ka];
      a.y = sa[ka + 1];
#pragma unroll
      for (int nt = 0; nt < 4; ++nt) {
        v2f b;
        b.x = sb[ka * SB_STRIDE + nt * 16];
        b.y = sb[(ka + 1) * SB_STRIDE + nt * 16];
        // 8 args: (neg_a, A, neg_b, B, c_mod, C, reuse_a, reuse_b)
        acc[nt] = __builtin_amdgcn_wmma_f32_16x16x4_f32(
            false, a, false, b, (short)0, acc[nt], false, false);
      }
    }

    wait_async0();    // my async copies into the next buffer are in LDS
    __syncthreads();  // everyone done computing `cur` + everyone's copies done
  }

  // Epilogue: optional bias, then store.
  float bv[4];
#pragma unroll
  for (int nt = 0; nt < 4; ++nt)
    bv[nt] = bias ? bias[col_blk + 64 * wn + nt * 16 + l] : 0.0f;

#pragma unroll
  for (int r = 0; r < 8; ++r) {
    const size_t ro =
        (size_t)(row_blk + 16 * wm + 8 * half + r) * N + col_blk + 64 * wn + l;
#pragma unroll
    for (int nt = 0; nt < 4; ++nt)
      C[ro + nt * 16] = acc[nt][r] + bv[nt];
  }
}

// ---------------------------------------------------------------------------
// Per-(b,h) fast-weight recurrence; the diagonal trick makes the scan purely
// elementwise, so all B*H = 131072 chains are independent. Coalesced xW reads
// (stride B*H between timesteps); xW (128MB) is L2-resident (192MB L2).
// ---------------------------------------------------------------------------
__global__ __launch_bounds__(256) void fw_scan(
    const float* __restrict__ xW,    // [T, B*H]
    const float* __restrict__ W_hh,  // [H, H] (only diagonal used)
    const float* __restrict__ b_ih,  // [H]
    const float* __restrict__ b_hh,  // [H]
    float* __restrict__ h_out) {     // [B*H]
  const int idx = blockIdx.x * blockDim.x + threadIdx.x;
  const int hi  = idx & (H_DIM - 1);

  const float w    = W_hh[(size_t)hi * H_DIM + hi];
  const float bias = b_ih[hi] + b_hh[hi];

  float h = 0.0f, Aacc = 0.0f, Macc = 0.0f;
  const float* xp     = xW + idx;
  const size_t stride = (size_t)B_DIM * H_DIM;

#pragma unroll 4
  for (int t = 0; t < T_DIM; ++t) {
    const float x = xp[(size_t)t * stride];
    Aacc = DECAY_LR * (Aacc + h * h);
    Macc = FAST_LR * (Macc + Aacc);
    const float pre = fmaf(w + Macc, h, bias + x);
    h = fmaxf(pre, 0.0f);
  }
  h_out[idx] = h;
}

// ---------------------------------------------------------------------------
extern "C" void kernel_launch(void* const* d_in, const int* in_sizes, int n_in,
                              void* d_out, int out_size, void* d_ws,
                              size_t ws_size, hipStream_t stream) {
  const float* inp  = (const float*)d_in[0];  // [T, B, I]
  const float* W_ih = (const float*)d_in[1];  // [I, H]
  const float* b_ih = (const float*)d_in[2];  // [H]
  const float* W_hh = (const float*)d_in[3];  // [H, H]
  const float* b_hh = (const float*)d_in[4];  // [H]
  const float* W_y  = (const float*)d_in[5];  // [H, O]
  const float* b_y  = (const float*)d_in[6];  // [O]
  float* out = (float*)d_out;                 // [B, O]

  float* xW     = (float*)d_ws;                        // T*B*H floats
  float* h_last = xW + (size_t)T_DIM * B_DIM * H_DIM;  // B*H floats

  // 1) Input projection GEMM: [32768, 512] x [512, 1024].
  {
    dim3 grid((T_DIM * B_DIM) / 32, H_DIM / 256);  // (1024, 4)
    fw_gemm_wmma<<<grid, 256, 0, stream>>>(inp, W_ih, nullptr, xW,
                                           T_DIM * B_DIM, I_DIM, H_DIM);
  }

  // 2) Elementwise 256-step recurrence.
  fw_scan<<<(B_DIM * H_DIM) / 256, 256, 0, stream>>>(xW, W_hh, b_ih, b_hh,
                                                     h_last);

  // 3) Output GEMM + bias: [128, 1024] x [1024, 512] + b_y.
  {
    dim3 grid(B_DIM / 32, O_DIM / 256);  // (4, 2)
    fw_gemm_wmma<<<grid, 256, 0, stream>>>(h_last, W_y, b_y, out,
                                           B_DIM, H_DIM, O_DIM);
  }
}